// GAT_GCN_78400333021316
// MI455X (gfx1250) — compile-verified
//
#include <hip/hip_runtime.h>
#include <hip/hip_bf16.h>
#include <math.h>

// ---------------- problem constants ----------------
#define N_NODES  50000
#define N_GRAPHS 128
#define HEADS    10
#define FXD      78
#define C780     780          // HEADS*FXD
#define E0       150000       // raw edges
#define ETOT     200000       // + self loops
#define EMB_L    128
#define EMB_CIN  1000

typedef __attribute__((ext_vector_type(2))) float v2f;
typedef __attribute__((ext_vector_type(8))) float v8f;
typedef __attribute__((ext_vector_type(4))) unsigned int u32x4;
typedef __attribute__((ext_vector_type(4))) int i32x4;
typedef __attribute__((ext_vector_type(8))) int i32x8;

static inline int cdiv(int a, int b) { return (a + b - 1) / b; }

__device__ __forceinline__ float atomicMaxF(float* addr, float val) {
  int* ai = (int*)addr;
  int old = __float_as_int(*addr);
  while (__int_as_float(old) < val) {
    int assumed = old;
    old = atomicCAS(ai, assumed, __float_as_int(val));
    if (old == assumed) break;
  }
  return __int_as_float(old);
}

// ---------------- TDM: async 2-D f32 tile load (global -> LDS) ----------------
// Packs D# group0/group1 per CDNA5 ISA 8.3/8.4 and issues TENSOR_LOAD_TO_LDS.
// tensor_d0/d1 are the REMAINING extents measured from the tile origin, so the
// TDM hardware zero-fills any out-of-range portion of the tile (our padding).
__device__ __forceinline__ void tdm_load_2d_f32(
    unsigned lds_off, const float* gptr,
    unsigned tensor_d0, unsigned tensor_d1,
    unsigned tile_d0, unsigned tile_d1, unsigned row_stride_elems)
{
  unsigned long long ga = (unsigned long long)(uintptr_t)gptr;
  u32x4 g0;
  g0.x = 0x1u;                                        // count=1 user descriptor
  g0.y = lds_off;                                     // lds_addr [63:32]
  g0.z = (unsigned)(ga & 0xFFFFFFFFu);                // global_addr[31:0]
  g0.w = (unsigned)((ga >> 32) & 0x01FFFFFFu) | (2u << 30);  // ga[56:32] | type=2
  i32x8 g1;
  g1[0] = (int)(2u << 16);                            // wg_mask=0, data_size=2 (4B)
  g1[1] = (int)((tensor_d0 & 0xFFFFu) << 16);         // atomic_addr=0 | td0[15:0]
  g1[2] = (int)(((tensor_d0 >> 16) & 0xFFFFu) | ((tensor_d1 & 0xFFFFu) << 16));
  g1[3] = (int)(((tensor_d1 >> 16) & 0xFFFFu) | ((tile_d0 & 0xFFFFu) << 16));
  g1[4] = (int)(tile_d1 & 0xFFFFu);                   // tile_dim1 | tile_dim2=0
  g1[5] = (int)row_stride_elems;                      // tensor_dim0_stride[31:0]
  g1[6] = 0;                                          // stride[47:32] | dim1_stride lo
  g1[7] = 0;
  i32x4 z4 = {0, 0, 0, 0};
#if __clang_major__ >= 23
  i32x8 z8 = {0, 0, 0, 0, 0, 0, 0, 0};
  __builtin_amdgcn_tensor_load_to_lds(g0, g1, z4, z4, z8, 0);
#else
  __builtin_amdgcn_tensor_load_to_lds(g0, g1, z4, z4, 0);
#endif
}

// ---------------- f32 WMMA GEMM: C = A[M,K] * B[K,N] (+bias)(+relu) -----------
// 64x64 block tile, 8 waves, each wave owns a 16x32 slab (2 WMMA accumulators).
// A/B tiles streamed by the Tensor Data Mover into double-buffered LDS,
// overlapped with WMMA compute; one barrier + s_wait_tensorcnt per K step.
#define BM 64
#define BN 64
#define BK 16

__global__ __launch_bounds__(256) void gemm_wmma(
    const float* __restrict__ A, const float* __restrict__ B,
    const float* __restrict__ bias, float* __restrict__ C,
    int M, int N, int K, int relu)
{
  __shared__ float As[2][BM][BK];
  __shared__ float Bs[2][BK][BN];

  const int tid  = threadIdx.x;
  const int lane = tid & 31;
  const int wave = tid >> 5;
  const int r16  = lane & 15;
  const int hi   = lane >> 4;
  const int wr   = (wave & 3) * 16;    // wave row offset inside block tile
  const int wc   = (wave >> 2) * 32;   // wave col offset (2 x 16-wide tiles)
  const int m0   = blockIdx.y * BM;
  const int n0   = blockIdx.x * BN;
  const bool leader = (wave == 0);     // one wave drives the TDM

  v8f acc0 = {0.f, 0.f, 0.f, 0.f, 0.f, 0.f, 0.f, 0.f};
  v8f acc1 = {0.f, 0.f, 0.f, 0.f, 0.f, 0.f, 0.f, 0.f};

  if (leader) {
    tdm_load_2d_f32((unsigned)(uintptr_t)&As[0][0][0], &A[(size_t)m0 * K],
                    (unsigned)K, (unsigned)(M - m0), BK, BM, (unsigned)K);
    tdm_load_2d_f32((unsigned)(uintptr_t)&Bs[0][0][0], &B[n0],
                    (unsigned)(N - n0), (unsigned)K, BN, BK, (unsigned)N);
  }

  int buf = 0;
  for (int kb = 0; kb < K; kb += BK) {
    if (leader) __builtin_amdgcn_s_wait_tensorcnt(0);  // current buffer landed
    __syncthreads();                                   // visible to all waves
    if (leader && (kb + BK < K)) {                     // stream next tiles async
      int kn = kb + BK;
      tdm_load_2d_f32((unsigned)(uintptr_t)&As[buf ^ 1][0][0],
                      &A[(size_t)m0 * K + kn],
                      (unsigned)(K - kn), (unsigned)(M - m0), BK, BM, (unsigned)K);
      tdm_load_2d_f32((unsigned)(uintptr_t)&Bs[buf ^ 1][0][0],
                      &B[(size_t)kn * N + n0],
                      (unsigned)(N - n0), (unsigned)(K - kn), BN, BK, (unsigned)N);
    }

#pragma unroll
    for (int kk = 0; kk < BK / 4; ++kk) {
      const int kf = kk * 4 + hi * 2;  // lanes 0-15: K base; lanes 16-31: +2
      v2f a  = { As[buf][wr + r16][kf],      As[buf][wr + r16][kf + 1] };
      v2f b0 = { Bs[buf][kf][wc + r16],      Bs[buf][kf + 1][wc + r16] };
      v2f b1 = { Bs[buf][kf][wc + 16 + r16], Bs[buf][kf + 1][wc + 16 + r16] };
      acc0 = __builtin_amdgcn_wmma_f32_16x16x4_f32(false, a, false, b0,
                                                   (short)0, acc0, false, false);
      acc1 = __builtin_amdgcn_wmma_f32_16x16x4_f32(false, a, false, b1,
                                                   (short)0, acc1, false, false);
    }
    buf ^= 1;
  }

  const int col0 = n0 + wc + r16;
  const int col1 = col0 + 16;
#pragma unroll
  for (int i = 0; i < 8; ++i) {
    int row = m0 + wr + hi * 8 + i;    // C/D VGPR i: rows i / i+8 by lane half
    if (row < M) {
      if (col0 < N) {
        float v = acc0[i];
        if (bias) v += bias[col0];
        if (relu) v = fmaxf(v, 0.f);
        C[(size_t)row * N + col0] = v;
      }
      if (col1 < N) {
        float v = acc1[i];
        if (bias) v += bias[col1];
        if (relu) v = fmaxf(v, 0.f);
        C[(size_t)row * N + col1] = v;
      }
    }
  }
}

// ---------------- elementwise / graph kernels ----------------
__global__ void fill_f32(float* __restrict__ p, int n, float v) {
  int i = blockIdx.x * blockDim.x + threadIdx.x;
  if (i < n) p[i] = v;
}

__global__ void att_scores(const float* __restrict__ H,
                           const float* __restrict__ att_src,
                           const float* __restrict__ att_dst,
                           float* __restrict__ a_s, float* __restrict__ a_d) {
  int idx = blockIdx.x * blockDim.x + threadIdx.x;
  if (idx >= N_NODES * HEADS) return;
  int n = idx / HEADS, h = idx % HEADS;
  const float* hr = H + (size_t)n * C780 + h * FXD;
  const float* ws = att_src + h * FXD;
  const float* wd = att_dst + h * FXD;
  float ss = 0.f, sd = 0.f;
  for (int f = 0; f < FXD; ++f) { float v = hr[f]; ss += v * ws[f]; sd += v * wd[f]; }
  a_s[idx] = ss;
  a_d[idx] = sd;
}

__device__ __forceinline__ void edge_endpoints(int e, const int* src, const int* dst,
                                               int& s, int& d) {
  if (e < E0) { s = src[e]; d = dst[e]; } else { s = e - E0; d = e - E0; }
}

__global__ void edge_logits_max(const int* __restrict__ src, const int* __restrict__ dst,
                                const float* __restrict__ a_s, const float* __restrict__ a_d,
                                float* __restrict__ e_val, float* __restrict__ m) {
  int idx = blockIdx.x * blockDim.x + threadIdx.x;
  if (idx >= ETOT * HEADS) return;
  int e = idx / HEADS, h = idx % HEADS;
  int s, d; edge_endpoints(e, src, dst, s, d);
  float v = a_s[s * HEADS + h] + a_d[d * HEADS + h];
  v = (v >= 0.f) ? v : 0.2f * v;          // leaky_relu 0.2
  e_val[idx] = v;
  atomicMaxF(&m[d * HEADS + h], v);
}

__global__ void edge_exp_sum(const int* __restrict__ src, const int* __restrict__ dst,
                             const float* __restrict__ e_val, const float* __restrict__ m,
                             float* __restrict__ ssum) {
  int idx = blockIdx.x * blockDim.x + threadIdx.x;
  if (idx >= ETOT * HEADS) return;
  int e = idx / HEADS, h = idx % HEADS;
  int s, d; edge_endpoints(e, src, dst, s, d);
  (void)s;
  atomicAdd(&ssum[d * HEADS + h], __expf(e_val[idx] - m[d * HEADS + h]));
}

__global__ void gat_aggregate(const int* __restrict__ src, const int* __restrict__ dst,
                              const float* __restrict__ H, const float* __restrict__ e_val,
                              const float* __restrict__ m, const float* __restrict__ ssum,
                              float* __restrict__ out) {
  int idx = blockIdx.x * blockDim.x + threadIdx.x;
  if (idx >= ETOT * C780) return;
  int e = idx / C780, f = idx % C780;
  int s, d; edge_endpoints(e, src, dst, s, d);
  int h = f / FXD;
  float alpha = __expf(e_val[e * HEADS + h] - m[d * HEADS + h]) /
                (ssum[d * HEADS + h] + 1e-16f);
  atomicAdd(&out[(size_t)d * C780 + f], H[(size_t)s * C780 + f] * alpha);
}

__global__ void bias_relu_rows(float* __restrict__ x, const float* __restrict__ bias,
                               int rows, int cols) {
  int idx = blockIdx.x * blockDim.x + threadIdx.x;
  if (idx >= rows * cols) return;
  x[idx] = fmaxf(x[idx] + bias[idx % cols], 0.f);
}

__global__ void degree_kernel(const int* __restrict__ src, const int* __restrict__ dst,
                              float* __restrict__ deg) {
  int e = blockIdx.x * blockDim.x + threadIdx.x;
  if (e >= ETOT) return;
  int s, d; edge_endpoints(e, src, dst, s, d);
  (void)s;
  atomicAdd(&deg[d], 1.f);
}

__global__ void dinv_kernel(const float* __restrict__ deg, float* __restrict__ dinv) {
  int n = blockIdx.x * blockDim.x + threadIdx.x;
  if (n >= N_NODES) return;
  float d = deg[n];
  dinv[n] = (d > 0.f) ? rsqrtf(d) : 0.f;
}

__global__ void gcn_aggregate(const int* __restrict__ src, const int* __restrict__ dst,
                              const float* __restrict__ H2, const float* __restrict__ dinv,
                              float* __restrict__ out) {
  int idx = blockIdx.x * blockDim.x + threadIdx.x;
  if (idx >= ETOT * C780) return;
  int e = idx / C780, f = idx % C780;
  int s, d; edge_endpoints(e, src, dst, s, d);
  float nrm = dinv[s] * dinv[d];
  atomicAdd(&out[(size_t)d * C780 + f], H2[(size_t)s * C780 + f] * nrm);
}

__global__ void pool_kernel(const float* __restrict__ h, const int* __restrict__ batch,
                            float* __restrict__ gmx, float* __restrict__ gsum) {
  int idx = blockIdx.x * blockDim.x + threadIdx.x;
  if (idx >= N_NODES * C780) return;
  int n = idx / C780, f = idx % C780;
  int g = batch[n];
  float v = h[idx];
  atomicMaxF(&gmx[g * C780 + f], v);
  atomicAdd(&gsum[g * C780 + f], v);
}

__global__ void count_kernel(const int* __restrict__ batch, float* __restrict__ cnt) {
  int n = blockIdx.x * blockDim.x + threadIdx.x;
  if (n >= N_NODES) return;
  atomicAdd(&cnt[batch[n]], 1.f);
}

__global__ void build_g(const float* __restrict__ gmx, const float* __restrict__ gsum,
                        const float* __restrict__ cnt, float* __restrict__ g) {
  int idx = blockIdx.x * blockDim.x + threadIdx.x;
  if (idx >= N_GRAPHS * C780) return;
  int b = idx / C780, f = idx % C780;
  float c = cnt[b];
  g[b * (2 * C780) + f]        = (c > 0.f) ? gmx[idx] : 0.f;
  g[b * (2 * C780) + C780 + f] = gsum[idx] / fmaxf(c, 1.f);
}

__global__ void embed_gather(const float* __restrict__ emb, const int* __restrict__ t2,
                             float* __restrict__ emb_mat) {
  int idx = blockIdx.x * blockDim.x + threadIdx.x;
  if (idx >= N_GRAPHS * EMB_CIN * EMB_L) return;
  int l = idx % EMB_L;
  int t = idx / EMB_L;         // b*1000 + ci
  emb_mat[idx] = emb[t2[t] * EMB_L + l];
}

// conv1d: emb_mat[B,1000,128] (*) Wc2[32,1000,8] -> xt2flat[B, 32*121]
__global__ __launch_bounds__(128) void conv2_kernel(
    const float* __restrict__ emb_mat, const float* __restrict__ Wc2,
    const float* __restrict__ bc2, float* __restrict__ out) {
  __shared__ float w[EMB_CIN * 8];              // 32 KB filter slice in LDS
  int b = blockIdx.x, co = blockIdx.y;
  for (int i = threadIdx.x; i < EMB_CIN * 8; i += 128)
    w[i] = Wc2[co * (EMB_CIN * 8) + i];
  __syncthreads();
  int l = threadIdx.x;
  if (l < 121) {
    float acc = bc2[co];
    const float* e = emb_mat + (size_t)b * EMB_CIN * EMB_L;
    for (int ci = 0; ci < EMB_CIN; ++ci) {
      const float* er = e + ci * EMB_L + l;
      const float* wr = w + ci * 8;
#pragma unroll
      for (int k = 0; k < 8; ++k) acc += er[k] * wr[k];
    }
    out[(size_t)b * (32 * 121) + co * 121 + l] = acc;
  }
}

// conv1d: target1[B,20,24] (*) Wc1[32,20,8] -> xt1flat[B, 32*17]
__global__ void conv1_kernel(const float* __restrict__ t1, const float* __restrict__ Wc1,
                             const float* __restrict__ bc1, float* __restrict__ out) {
  int idx = blockIdx.x * blockDim.x + threadIdx.x;
  if (idx >= N_GRAPHS * 32 * 17) return;
  int b = idx / (32 * 17);
  int r = idx % (32 * 17);
  int co = r / 17, l = r % 17;
  float acc = bc1[co];
  const float* x = t1 + b * (20 * 24);
  const float* w = Wc1 + co * (20 * 8);
  for (int ci = 0; ci < 20; ++ci)
#pragma unroll
    for (int k = 0; k < 8; ++k) acc += x[ci * 24 + l + k] * w[ci * 8 + k];
  out[b * (32 * 17) + co * 17 + l] = acc;
}

__global__ void concat_kernel(const float* __restrict__ g2, const float* __restrict__ xt1,
                              const float* __restrict__ xt2, float* __restrict__ xc) {
  int idx = blockIdx.x * blockDim.x + threadIdx.x;
  if (idx >= N_GRAPHS * 384) return;
  int b = idx / 384, j = idx % 384;
  float v;
  if (j < 128)      v = g2 [b * 128 + j];
  else if (j < 256) v = xt1[b * 128 + j - 128];
  else              v = xt2[b * 128 + j - 256];
  xc[idx] = v;
}

__global__ void final_fc(const float* __restrict__ X, const float* __restrict__ W,
                         const float* __restrict__ b, float* __restrict__ out) {
  int i = blockIdx.x * blockDim.x + threadIdx.x;
  if (i >= N_GRAPHS) return;
  float acc = b[0];
  for (int k = 0; k < 512; ++k) acc += X[i * 512 + k] * W[k];
  out[i] = acc;
}

// ---------------- host launcher ----------------
static inline void launch_gemm(const float* A, const float* B, const float* bias,
                               float* C, int M, int N, int K, int relu,
                               hipStream_t s) {
  dim3 grid(cdiv(N, BN), cdiv(M, BM));
  gemm_wmma<<<grid, 256, 0, s>>>(A, B, bias, C, M, N, K, relu);
}

extern "C" void kernel_launch(void* const* d_in, const int* in_sizes, int n_in,
                              void* d_out, int out_size, void* d_ws, size_t ws_size,
                              hipStream_t stream) {
  (void)in_sizes; (void)n_in; (void)out_size; (void)ws_size;

  const float* x       = (const float*)d_in[0];
  const int*   ei      = (const int*)  d_in[1];
  const int*   batch   = (const int*)  d_in[2];
  const float* target1 = (const float*)d_in[3];
  const int*   target2 = (const int*)  d_in[4];
  const float* W_gat   = (const float*)d_in[5];
  const float* att_src = (const float*)d_in[6];
  const float* att_dst = (const float*)d_in[7];
  const float* b_gat   = (const float*)d_in[8];
  const float* W_gcn   = (const float*)d_in[9];
  const float* b_gcn   = (const float*)d_in[10];
  const float* W_fcg1  = (const float*)d_in[11];
  const float* b_fcg1  = (const float*)d_in[12];
  const float* W_fcg2  = (const float*)d_in[13];
  const float* b_fcg2  = (const float*)d_in[14];
  const float* emb     = (const float*)d_in[15];
  const float* Wc2     = (const float*)d_in[16];
  const float* bc2     = (const float*)d_in[17];
  const float* W_fc2xt = (const float*)d_in[18];
  const float* b_fc2xt = (const float*)d_in[19];
  const float* Wc1     = (const float*)d_in[20];
  const float* bc1     = (const float*)d_in[21];
  const float* W_fc1xt = (const float*)d_in[22];
  const float* b_fc1xt = (const float*)d_in[23];
  const float* W_fc1   = (const float*)d_in[24];
  const float* b_fc1   = (const float*)d_in[25];
  const float* W_fc2   = (const float*)d_in[26];
  const float* b_fc2   = (const float*)d_in[27];
  const float* W_out   = (const float*)d_in[28];
  const float* b_out   = (const float*)d_in[29];
  float* out = (float*)d_out;

  const int* srcA = ei;           // row 0
  const int* dstA = ei + E0;      // row 1

  // ---- workspace bump allocator (with aliasing for big buffers) ----
  char* ws = (char*)d_ws;
  size_t off = 0;
  auto alloc = [&](size_t bytes) -> float* {
    float* p = (float*)(ws + off);
    off += (bytes + 255) & ~(size_t)255;
    return p;
  };
  const size_t BIG = (size_t)N_NODES * C780 * sizeof(float);   // 156 MB
  float* H1     = alloc(BIG);                   // GAT GEMM out (pre-softmax features)
  float* GATOUT = alloc(BIG);                   // GAT aggregated output
  float* H2     = alloc(BIG);                   // GCN GEMM out
  float* a_s    = alloc((size_t)N_NODES * HEADS * 4);
  float* a_d    = alloc((size_t)N_NODES * HEADS * 4);
  float* e_val  = alloc((size_t)ETOT * HEADS * 4);
  float* mMax   = alloc((size_t)N_NODES * HEADS * 4);
  float* sSum   = alloc((size_t)N_NODES * HEADS * 4);
  float* deg    = alloc((size_t)N_NODES * 4);
  float* dinv   = alloc((size_t)N_NODES * 4);
  float* cnt    = alloc((size_t)N_GRAPHS * 4);
  float* gmx    = alloc((size_t)N_GRAPHS * C780 * 4);
  float* gsum   = alloc((size_t)N_GRAPHS * C780 * 4);
  float* gfeat  = alloc((size_t)N_GRAPHS * 2 * C780 * 4);
  float* g1     = alloc((size_t)N_GRAPHS * 1500 * 4);
  float* g2     = alloc((size_t)N_GRAPHS * 128 * 4);
  float* xt2f   = alloc((size_t)N_GRAPHS * 32 * 121 * 4);
  float* xt2    = alloc((size_t)N_GRAPHS * 128 * 4);
  float* xt1f   = alloc((size_t)N_GRAPHS * 32 * 17 * 4);
  float* xt1    = alloc((size_t)N_GRAPHS * 128 * 4);
  float* xc     = alloc((size_t)N_GRAPHS * 384 * 4);
  float* xc1    = alloc((size_t)N_GRAPHS * 1024 * 4);
  float* xc2    = alloc((size_t)N_GRAPHS * 512 * 4);
  float* GCNOUT  = H1;      // safe: H1 dead after gat_aggregate
  float* emb_mat = GATOUT;  // safe: GATOUT dead after GCN GEMM

  const int TB = 256;

  // ---- GAT ----
  launch_gemm(x, W_gat, nullptr, H1, N_NODES, C780, FXD, 0, stream);
  att_scores<<<cdiv(N_NODES * HEADS, TB), TB, 0, stream>>>(H1, att_src, att_dst, a_s, a_d);
  fill_f32<<<cdiv(N_NODES * HEADS, TB), TB, 0, stream>>>(mMax, N_NODES * HEADS, -3.0e38f);
  hipMemsetAsync(sSum, 0, (size_t)N_NODES * HEADS * 4, stream);
  hipMemsetAsync(GATOUT, 0, BIG, stream);
  hipMemsetAsync(deg, 0, (size_t)N_NODES * 4, stream);
  edge_logits_max<<<cdiv(ETOT * HEADS, TB), TB, 0, stream>>>(srcA, dstA, a_s, a_d, e_val, mMax);
  edge_exp_sum<<<cdiv(ETOT * HEADS, TB), TB, 0, stream>>>(srcA, dstA, e_val, mMax, sSum);
  gat_aggregate<<<cdiv(ETOT * C780, TB), TB, 0, stream>>>(srcA, dstA, H1, e_val, mMax, sSum, GATOUT);
  bias_relu_rows<<<cdiv(N_NODES * C780, TB), TB, 0, stream>>>(GATOUT, b_gat, N_NODES, C780);

  // ---- GCN ----
  hipMemsetAsync(GCNOUT, 0, BIG, stream);                 // aliases H1 (now dead)
  degree_kernel<<<cdiv(ETOT, TB), TB, 0, stream>>>(srcA, dstA, deg);
  dinv_kernel<<<cdiv(N_NODES, TB), TB, 0, stream>>>(deg, dinv);
  launch_gemm(GATOUT, W_gcn, nullptr, H2, N_NODES, C780, C780, 0, stream);
  gcn_aggregate<<<cdiv(ETOT * C780, TB), TB, 0, stream>>>(srcA, dstA, H2, dinv, GCNOUT);
  bias_relu_rows<<<cdiv(N_NODES * C780, TB), TB, 0, stream>>>(GCNOUT, b_gcn, N_NODES, C780);

  // ---- pooling ----
  fill_f32<<<cdiv(N_GRAPHS * C780, TB), TB, 0, stream>>>(gmx, N_GRAPHS * C780, -3.0e38f);
  hipMemsetAsync(gsum, 0, (size_t)N_GRAPHS * C780 * 4, stream);
  hipMemsetAsync(cnt, 0, (size_t)N_GRAPHS * 4, stream);
  pool_kernel<<<cdiv(N_NODES * C780, TB), TB, 0, stream>>>(GCNOUT, batch, gmx, gsum);
  count_kernel<<<cdiv(N_NODES, TB), TB, 0, stream>>>(batch, cnt);
  build_g<<<cdiv(N_GRAPHS * C780, TB), TB, 0, stream>>>(gmx, gsum, cnt, gfeat);

  // ---- graph FC head ----
  launch_gemm(gfeat, W_fcg1, b_fcg1, g1, N_GRAPHS, 1500, 1560, 1, stream);
  launch_gemm(g1, W_fcg2, b_fcg2, g2, N_GRAPHS, 128, 1500, 0, stream);

  // ---- protein branch 2 (embedding + conv1d) ----
  embed_gather<<<cdiv(N_GRAPHS * EMB_CIN * EMB_L, TB), TB, 0, stream>>>(emb, target2, emb_mat);
  {
    dim3 grid(N_GRAPHS, 32);
    conv2_kernel<<<grid, 128, 0, stream>>>(emb_mat, Wc2, bc2, xt2f);
  }
  launch_gemm(xt2f, W_fc2xt, b_fc2xt, xt2, N_GRAPHS, 128, 32 * 121, 0, stream);

  // ---- protein branch 1 ----
  conv1_kernel<<<cdiv(N_GRAPHS * 32 * 17, TB), TB, 0, stream>>>(target1, Wc1, bc1, xt1f);
  launch_gemm(xt1f, W_fc1xt, b_fc1xt, xt1, N_GRAPHS, 128, 544, 0, stream);

  // ---- combined head ----
  concat_kernel<<<cdiv(N_GRAPHS * 384, TB), TB, 0, stream>>>(g2, xt1, xt2, xc);
  launch_gemm(xc, W_fc1, b_fc1, xc1, N_GRAPHS, 1024, 384, 1, stream);
  launch_gemm(xc1, W_fc2, b_fc2, xc2, N_GRAPHS, 512, 1024, 1, stream);
  final_fc<<<cdiv(N_GRAPHS, 128), 128, 0, stream>>>(xc2, W_out, b_out, out);
}